// Model_44255343018774
// MI455X (gfx1250) — compile-verified
//
#include <hip/hip_runtime.h>

typedef float v2f  __attribute__((ext_vector_type(2)));
typedef float v8f  __attribute__((ext_vector_type(8)));
typedef unsigned int u32x4 __attribute__((ext_vector_type(4)));
typedef int   i32x4 __attribute__((ext_vector_type(4)));
typedef int   i32x8 __attribute__((ext_vector_type(8)));

namespace {
constexpr int BB    = 32;
constexpr int LL    = 2880;
constexpr int CC    = 862;
constexpr int PRED  = 720;
constexpr int SEG   = 48;
constexpr int NMAP  = 4;
constexpr int KW    = 16;
constexpr int NIN   = 60;    // L / SEG
constexpr int NOUT  = 15;    // PRED / SEG
constexpr int CONVD = 359;   // (L-K)/8 + 1

constexpr int CT    = 16;    // channels per workgroup
constexpr int NTHR  = 256;   // 8 waves
constexpr int NBLKC = (CC + CT - 1) / CT;   // 54

// LDS layout (floats).
// X tile stored row-major [l][c] with pitch 17 — produced directly by the
// TDM using pad_interval=16 DWORDs / pad_amount=1 DWORD. Pitch 17 (odd) makes
// the WMMA A-loads ((i*48+s)*17 + c) conflict-free across lanes.
constexpr int XP      = CT + 1;          // 17
constexpr int WPITCH  = 16 * NIN;        // 960: W_eff with o padded 15->16 (zeros)
constexpr int SROW    = 17;              // stage row stride per s
constexpr int SPITCH  = SEG * SROW + 1;  // 817
constexpr int XOFF    = 0;
constexpr int WOFF    = XOFF + LL * XP;              // 48960
constexpr int SOFF    = WOFF + CT * WPITCH;          // 64320
constexpr int REDOFF  = SOFF + CT * SPITCH;          // 77392 (16ch x 16slots x 4)
constexpr int MEANOFF = REDOFF + CT * 16 * NMAP;     // 78416
constexpr int STDOFF  = MEANOFF + CT;
constexpr int GATEOFF = STDOFF + CT;                 // 16 x 4
constexpr int BEFFOFF = GATEOFF + CT * NMAP;         // 16 x 15
constexpr int LDS_FLOATS = BEFFOFF + CT * NOUT;      // 78752 -> 315008 bytes
constexpr size_t LDS_BYTES = (size_t)LDS_FLOATS * sizeof(float);
} // namespace

__global__ __launch_bounds__(NTHR)
void fused_moe_forecast(const float* __restrict__ x,
                        const float* __restrict__ conv_w,
                        const float* __restrict__ conv_b,
                        const float* __restrict__ gate_w,
                        const float* __restrict__ gate_b,
                        const float* __restrict__ map_w,
                        const float* __restrict__ map_b,
                        float* __restrict__ out)
{
    extern __shared__ float lds[];
    const int tid   = threadIdx.x;
    const int b     = blockIdx.y;
    const int cbase = blockIdx.x * CT;
    const int ccl   = tid & 15;          // channel-in-tile for phases 1-2
    const int slot  = tid >> 4;          // 0..15

    // ---------------- Phase 0: TDM load x[b, :, cbase:cbase+16] -> lds[l][c] ------------
    // One TENSOR_LOAD_TO_LDS for the whole 2880x16 fp32 tile. The D# pad feature
    // (pad every 16 DWORDs by 1 DWORD) yields LDS row pitch 17. tensor_dim0 is the
    // number of valid columns remaining from the tile start, so the TDM zero-fills
    // the tail channels of the last block for us.
    if (tid < 32) {    // wave 0 only; wave-uniform -> other waves skip via execz branch
        const unsigned long long gaddr =
            (unsigned long long)(x + (size_t)b * LL * CC + cbase);
        const unsigned ldsbase = __builtin_amdgcn_groupstaticsize() + XOFF * 4u;
        const int dim0 = CC - cbase;     // valid columns from tile start (>= 14)

        u32x4 g0;
        g0[0] = 1u;                                       // count=1 (valid), user mode
        g0[1] = ldsbase;                                  // lds_addr (bytes)
        g0[2] = (unsigned)(gaddr & 0xFFFFFFFFull);        // global_addr[31:0]
        g0[3] = (unsigned)((gaddr >> 32) & 0x1FFFFFFull)  // global_addr[56:32]
              | (2u << 30);                               // type = 2 ("image")

        i32x8 g1;
        g1[0] = (2 << 16)        // data_size = 4 bytes
              | (1 << 20)        // pad_enable
              | (3 << 22)        // pad_interval: 16 DWORDs
              | (0 << 25);       // pad_amount:   1 DWORD  -> pitch 17 floats
        g1[1] = (dim0 & 0xFFFF) << 16;                    // tensor_dim0[15:0]
        g1[2] = ((dim0 >> 16) & 0xFFFF) | ((LL & 0xFFFF) << 16); // dim0 hi | dim1 lo
        g1[3] = ((LL >> 16) & 0xFFFF) | (CT << 16);       // dim1 hi | tile_dim0 = 16
        g1[4] = (LL & 0xFFFF);                            // tile_dim1 = 2880, tile_dim2 = 0
        g1[5] = CC;                                       // tensor_dim0_stride = 862
        g1[6] = 0;
        g1[7] = 0;

        i32x4 gz4 = {0, 0, 0, 0};                         // 2-D tensor: groups 2/3 unused
        i32x8 gz8 = {0, 0, 0, 0, 0, 0, 0, 0};
        __builtin_amdgcn_tensor_load_to_lds(g0, g1, gz4, gz4, gz8, 0);
        __builtin_amdgcn_s_wait_tensorcnt(0);
    }
    __syncthreads();

    // ---------------- Phase 1: per-channel mean / stdev (biased var + 1e-10) ------------
    {
        const int l0 = slot * (LL / 16);       // 180 elems per (ch, slot)
        float s = 0.f, s2 = 0.f;
        #pragma unroll 4
        for (int j = 0; j < LL / 16; ++j) {
            float v = lds[XOFF + (l0 + j) * XP + ccl];
            s += v; s2 += v * v;
        }
        lds[REDOFF + (ccl * 16 + slot) * 2 + 0] = s;
        lds[REDOFF + (ccl * 16 + slot) * 2 + 1] = s2;
    }
    __syncthreads();
    if (tid < CT) {
        float s = 0.f, s2 = 0.f;
        #pragma unroll
        for (int j = 0; j < 16; ++j) {
            s  += lds[REDOFF + (tid * 16 + j) * 2 + 0];
            s2 += lds[REDOFF + (tid * 16 + j) * 2 + 1];
        }
        const float mean = s / (float)LL;
        float var = s2 / (float)LL - mean * mean;
        var = fmaxf(var, 0.f);
        const float sd = sqrtf(var + 1e-10f);
        lds[MEANOFF + tid] = mean;
        lds[STDOFF + tid]  = sd;
    }
    __syncthreads();

    // ---------------- Phase 1b: normalize in place --------------------------------------
    {
        const float mean = lds[MEANOFF + ccl];
        const float rstd = 1.f / lds[STDOFF + ccl];
        const int l0 = slot * (LL / 16);
        #pragma unroll 4
        for (int j = 0; j < LL / 16; ++j) {
            const int a = XOFF + (l0 + j) * XP + ccl;
            lds[a] = (lds[a] - mean) * rstd;
        }
    }
    __syncthreads();

    // ---------------- Phase 2: depthwise conv -> gate dots -> softmax -------------------
    {
        const int csafe = min(cbase + ccl, CC - 1);
        float wreg[KW];
        #pragma unroll
        for (int k = 0; k < KW; ++k) wreg[k] = conv_w[csafe * KW + k];
        const float cb = conv_b[csafe];
        float g0 = 0.f, g1 = 0.f, g2 = 0.f, g3 = 0.f;
        for (int d = slot; d < CONVD; d += 16) {
            float acc = cb;
            const int l0 = d * 8;
            #pragma unroll
            for (int k = 0; k < KW; ++k)
                acc += lds[XOFF + (l0 + k) * XP + ccl] * wreg[k];
            g0 += acc * gate_w[0 * CONVD + d];
            g1 += acc * gate_w[1 * CONVD + d];
            g2 += acc * gate_w[2 * CONVD + d];
            g3 += acc * gate_w[3 * CONVD + d];
        }
        float* red = &lds[REDOFF + (ccl * 16 + slot) * NMAP];
        red[0] = g0; red[1] = g1; red[2] = g2; red[3] = g3;
    }
    __syncthreads();
    if (tid < CT) {
        float g[NMAP];
        #pragma unroll
        for (int m = 0; m < NMAP; ++m) {
            float s = gate_b[m];
            for (int j = 0; j < 16; ++j)
                s += lds[REDOFF + (tid * 16 + j) * NMAP + m];
            g[m] = s;
        }
        float mx = fmaxf(fmaxf(g[0], g[1]), fmaxf(g[2], g[3]));
        float den = 0.f;
        #pragma unroll
        for (int m = 0; m < NMAP; ++m) { g[m] = __expf(g[m] - mx); den += g[m]; }
        const float rden = 1.f / den;
        #pragma unroll
        for (int m = 0; m < NMAP; ++m) lds[GATEOFF + tid * NMAP + m] = g[m] * rden;
    }
    __syncthreads();

    // ---------------- Phase 3: W_eff[c][o][i] (o padded to 16 with zeros) + b_eff -------
    for (int idx = tid; idx < CT * 16 * NIN; idx += NTHR) {
        const int c  = idx / (16 * NIN);
        const int r  = idx - c * (16 * NIN);
        const int o  = r / NIN;
        const int i  = r - o * NIN;
        float v = 0.f;
        if (o < NOUT) {
            #pragma unroll
            for (int m = 0; m < NMAP; ++m)
                v += lds[GATEOFF + c * NMAP + m] * map_w[(m * NOUT + o) * NIN + i];
        }
        lds[WOFF + c * WPITCH + o * NIN + i] = v;
    }
    for (int idx = tid; idx < CT * NOUT; idx += NTHR) {
        const int c = idx / NOUT;
        const int o = idx - c * NOUT;
        float v = 0.f;
        #pragma unroll
        for (int m = 0; m < NMAP; ++m)
            v += lds[GATEOFF + c * NMAP + m] * map_b[m * NOUT + o];
        lds[BEFFOFF + c * NOUT + o] = v;
    }
    __syncthreads();

    // ---------------- Phase 4: per-channel 48x60 @ 60x16 GEMM via f32 WMMA --------------
    // D[s,o] = sum_i xseg[s,i] * W_eff[o,i];  A:16(s)x4(i), B:4(i)x16(o), C/D:16x16 f32
    {
        const int wv   = tid >> 5;
        const int lane = tid & 31;
        const int m16  = lane & 15;
        const int hi   = lane >> 4;
        #pragma unroll
        for (int ci = 0; ci < 2; ++ci) {
            const int c      = wv * 2 + ci;
            const float mean = lds[MEANOFF + c];
            const float sd   = lds[STDOFF + c];
            v8f acc0 = {0.f,0.f,0.f,0.f,0.f,0.f,0.f,0.f};
            v8f acc1 = acc0, acc2 = acc0;
            for (int kb = 0; kb < NIN / 4; ++kb) {          // 15 K-steps of 4
                const int i0 = kb * 4 + hi * 2;
                // B tile: lane(lo/hi) holds K = {i0, i0+1}, N = m16 (i-contiguous -> b64)
                v2f bm = *(const v2f*)&lds[WOFF + c * WPITCH + m16 * NIN + i0];
                // A tiles: lane holds row s = 16t + m16, K = {i0, i0+1}
                v2f a0, a1, a2;
                a0.x = lds[XOFF + ((i0    ) * SEG +  0 + m16) * XP + c];
                a0.y = lds[XOFF + ((i0 + 1) * SEG +  0 + m16) * XP + c];
                a1.x = lds[XOFF + ((i0    ) * SEG + 16 + m16) * XP + c];
                a1.y = lds[XOFF + ((i0 + 1) * SEG + 16 + m16) * XP + c];
                a2.x = lds[XOFF + ((i0    ) * SEG + 32 + m16) * XP + c];
                a2.y = lds[XOFF + ((i0 + 1) * SEG + 32 + m16) * XP + c];
                acc0 = __builtin_amdgcn_wmma_f32_16x16x4_f32(false, a0, false, bm,
                                                             (short)0, acc0, false, false);
                acc1 = __builtin_amdgcn_wmma_f32_16x16x4_f32(false, a1, false, bm,
                                                             (short)0, acc1, false, false);
                acc2 = __builtin_amdgcn_wmma_f32_16x16x4_f32(false, a2, false, bm,
                                                             (short)0, acc2, false, false);
            }
            // Epilogue: D element (vgpr r, lane) -> s = 16t + r + 8*hi, o = m16
            const int   o  = m16;
            const float be = (o < NOUT) ? lds[BEFFOFF + c * NOUT + o] : 0.f;
            #pragma unroll
            for (int r = 0; r < 8; ++r) {
                const int sl = r + 8 * hi;
                if (o < NOUT) {
                    lds[SOFF + c * SPITCH + (sl     ) * SROW + o] = (acc0[r] + be) * sd + mean;
                    lds[SOFF + c * SPITCH + (sl + 16) * SROW + o] = (acc1[r] + be) * sd + mean;
                    lds[SOFF + c * SPITCH + (sl + 32) * SROW + o] = (acc2[r] + be) * sd + mean;
                }
            }
        }
    }
    __syncthreads();

    // ---------------- Phase 5: coalesced write-out out[b, p, c] -------------------------
    {
        const int c = cbase + ccl;
        const size_t obase = (size_t)b * PRED * CC;
        #pragma unroll 4
        for (int j = 0; j < PRED / 16; ++j) {
            const int p = slot + j * 16;
            const int o = p / SEG;
            const int s = p - o * SEG;
            const float v = lds[SOFF + ccl * SPITCH + s * SROW + o];
            if (c < CC)
                out[obase + (size_t)p * CC + c] = v;
        }
    }
}

extern "C" void kernel_launch(void* const* d_in, const int* in_sizes, int n_in,
                              void* d_out, int out_size, void* d_ws, size_t ws_size,
                              hipStream_t stream) {
    const float* x      = (const float*)d_in[0];
    const float* conv_w = (const float*)d_in[1];
    const float* conv_b = (const float*)d_in[2];
    const float* gate_w = (const float*)d_in[3];
    const float* gate_b = (const float*)d_in[4];
    const float* map_w  = (const float*)d_in[5];
    const float* map_b  = (const float*)d_in[6];
    float* out = (float*)d_out;

    (void)in_sizes; (void)n_in; (void)out_size; (void)d_ws; (void)ws_size;

    // Allow ~308 KB dynamic LDS (WGP supports up to 320 KB per workgroup).
    (void)hipFuncSetAttribute((const void*)fused_moe_forecast,
                              hipFuncAttributeMaxDynamicSharedMemorySize, (int)LDS_BYTES);

    dim3 grid(NBLKC, BB, 1);
    dim3 block(NTHR, 1, 1);
    fused_moe_forecast<<<grid, block, LDS_BYTES, stream>>>(
        x, conv_w, conv_b, gate_w, gate_b, map_w, map_b, out);
}